// QuantumGenerator_180388627018
// MI455X (gfx1250) — compile-verified
//
#include <hip/hip_runtime.h>
#include <math.h>

#define NQ    11
#define DIM   (1 << NQ)          // 2048 amplitudes
#define NC    3
#define NROT  6
#define BATCH 512
#define NANG  (NC * NROT * NQ)   // 198 angles per batch element
#define NLIN  (NC * NQ * NROT)   // 198 linear outputs (C*66)
#define NOUT  (2 * NQ)           // 22 expectation values

typedef __attribute__((ext_vector_type(2))) float v2f;
typedef __attribute__((ext_vector_type(8))) float v8f;

// ---------------- complex helpers ----------------
__device__ __forceinline__ float2 cmul(float2 a, float2 b) {
    return make_float2(a.x * b.x - a.y * b.y, a.x * b.y + a.y * b.x);
}
__device__ __forceinline__ float2 cadd(float2 a, float2 b) {
    return make_float2(a.x + b.x, a.y + b.y);
}
// 2x2 complex matmul C = A*B; matrices stored {m00,m01,m10,m11}
__device__ __forceinline__ void mat2mul(const float2* A, const float2* B, float2* C) {
    C[0] = cadd(cmul(A[0], B[0]), cmul(A[1], B[2]));
    C[1] = cadd(cmul(A[0], B[1]), cmul(A[1], B[3]));
    C[2] = cadd(cmul(A[2], B[0]), cmul(A[3], B[2]));
    C[3] = cadd(cmul(A[2], B[1]), cmul(A[3], B[3]));
}
__device__ __forceinline__ void ry_mat(float t, float2* M) {
    float s, c;
    __sincosf(0.5f * t, &s, &c);
    M[0] = make_float2(c, 0.f);  M[1] = make_float2(-s, 0.f);
    M[2] = make_float2(s, 0.f);  M[3] = make_float2(c, 0.f);
}
__device__ __forceinline__ void rz_mat(float t, float2* M) {
    float s, c;
    __sincosf(0.5f * t, &s, &c);
    M[0] = make_float2(c, -s);       M[1] = make_float2(0.f, 0.f);
    M[2] = make_float2(0.f, 0.f);    M[3] = make_float2(c, s);
}

// ============================================================
// Kernel 1: angle GEMM via V_WMMA_F32_16X16X4_F32
//   out[b][n] = sum_k x[b][k] * W[n][k] + bias[n]
//   (W flattened [3*66][11], n = c*66 + r*11 + q)
// One wave (32 lanes) per 16x16 output tile; K = 11 padded to 12.
// Fragment loads use clamped addresses + value masking so all
// global_load_b32 are unconditional (no EXEC-mask branching).
// ============================================================
__global__ __launch_bounds__(32) void angles_wmma_kernel(
    const float* __restrict__ x,     // [512][11]
    const float* __restrict__ W,     // [198][11]
    const float* __restrict__ bias,  // [198]
    float* __restrict__ out)         // [512][198]
{
    const int m0   = blockIdx.x * 16;       // batch-row tile
    const int n0   = blockIdx.y * 16;       // output-col tile
    const int lane = threadIdx.x;           // 0..31 (wave32)
    const int half = lane >> 4;             // 0: lanes 0-15, 1: lanes 16-31
    const int l15  = lane & 15;
    const int m    = m0 + l15;              // A-fragment row (always < 512)
    const int n    = n0 + l15;              // B/D-fragment column

    // Clamp B row address so every load is in-bounds; out-of-range columns
    // accumulate into lanes whose results are never stored.
    const int nc = (n < NLIN) ? n : (NLIN - 1);
    const float* __restrict__ xrow = x + m * NQ;
    const float* __restrict__ wrow = W + nc * NQ;

    v8f acc = {};
    #pragma unroll
    for (int k0 = 0; k0 < 12; k0 += 4) {
        v2f a, b;
        #pragma unroll
        for (int v = 0; v < 2; ++v) {
            const int k = k0 + v + 2 * half;      // ISA 16x4 f32 A layout
            const int   kc    = (k < NQ) ? k : 0; // valid address always
            const float kmask = (k < NQ) ? 1.f : 0.f;
            a[v] = xrow[kc] * kmask;              // zero K-padding by value
            b[v] = wrow[kc] * kmask;
        }
        // D = A(16x4) * B(4x16) + C  -> v_wmma_f32_16x16x4_f32
        acc = __builtin_amdgcn_wmma_f32_16x16x4_f32(
            /*neg_a=*/false, a, /*neg_b=*/false, b,
            /*c_mod=*/(short)0, acc, /*reuse_a=*/false, /*reuse_b=*/false);
    }

    if (n < NLIN) {
        const float bv = bias[n];
        #pragma unroll
        for (int j = 0; j < 8; ++j) {
            const int row = m0 + j + 8 * half;    // ISA 16x16 f32 C/D layout
            out[row * NLIN + n] = acc[j] + bv;
        }
    }
}

// ============================================================
// Kernel 2: statevector simulation, one workgroup per batch row.
// State (2048 complex64 = 16KB) lives in LDS. Gates fused:
//   U1  = RZ(a3)*RY(a2)*RZ(a1)*RY(a0)   (per qubit, per circuit)
//   Uc  = RZ(a5)*RY(a4)                 (controlled, per qubit)
// ============================================================
__global__ __launch_bounds__(256) void qsim_kernel(
    const float* __restrict__ x,       // [512][11]   (inline-angle fallback)
    const float* __restrict__ W,       // [198][11]
    const float* __restrict__ bias,    // [198]
    const float* __restrict__ ang_pre, // [512][198] from WMMA kernel, or null
    float* __restrict__ out)           // [512][22]
{
    __shared__ float2 st[DIM];                 // 16 KB statevector
    __shared__ float  ang[NANG];               // 198 angles
    __shared__ float2 u1[NC][NQ][4];           // fused 1q gates
    __shared__ float2 uc[NC][NQ][4];           // fused controlled gates
    __shared__ float  acc[NOUT];               // reduction targets

    const int tid = threadIdx.x;
    const int bb  = blockIdx.x;

    // ---- angles: precomputed (WMMA path) or inline dot products ----
    if (tid < NANG) {
        float a;
        if (ang_pre) {
            a = ang_pre[bb * NANG + tid];
        } else {
            a = bias[tid];
            #pragma unroll
            for (int i = 0; i < NQ; ++i)
                a += W[tid * NQ + i] * x[bb * NQ + i];
        }
        ang[tid] = a;
    }
    // ---- init state |0...0>, zero reduction accumulators ----
    for (int i = tid; i < DIM; i += 256) st[i] = make_float2(0.f, 0.f);
    if (tid == 0) st[0] = make_float2(1.f, 0.f);
    if (tid < NOUT) acc[tid] = 0.f;
    __syncthreads();

    // ---- build fused 2x2 gate matrices (33 threads) ----
    if (tid < NC * NQ) {
        const int c = tid / NQ, q = tid % NQ;
        const float* a = &ang[c * (NROT * NQ)];
        float2 A[4], B[4], T[4], U[4];
        ry_mat(a[0 * NQ + q], A);            // RY(a0)
        rz_mat(a[1 * NQ + q], B);            // RZ(a1)
        mat2mul(B, A, T);                    // RZ*RY
        ry_mat(a[2 * NQ + q], A);            // RY(a2)
        mat2mul(A, T, U);                    // RY*RZ*RY
        rz_mat(a[3 * NQ + q], B);            // RZ(a3)
        mat2mul(B, U, T);                    // full fused 1q gate
        #pragma unroll
        for (int k = 0; k < 4; ++k) u1[c][q][k] = T[k];

        ry_mat(a[4 * NQ + q], A);            // RY(a4)
        rz_mat(a[5 * NQ + q], B);            // RZ(a5)
        mat2mul(B, A, T);                    // fused controlled gate body
        #pragma unroll
        for (int k = 0; k < 4; ++k) uc[c][q][k] = T[k];
    }
    __syncthreads();

    // ---- circuit layers ----
    for (int c = 0; c < NC; ++c) {
        // fused single-qubit gates (1024 pairs each, 4 per thread)
        for (int q = 0; q < NQ; ++q) {
            const float2 g00 = u1[c][q][0], g01 = u1[c][q][1];
            const float2 g10 = u1[c][q][2], g11 = u1[c][q][3];
            const int bp = NQ - 1 - q, str = 1 << bp;
            for (int p = tid; p < DIM / 2; p += 256) {
                const int i0 = ((p >> bp) << (bp + 1)) | (p & (str - 1));
                const int i1 = i0 | str;
                const float2 s0 = st[i0], s1 = st[i1];
                st[i0] = cadd(cmul(g00, s0), cmul(g01, s1));
                st[i1] = cadd(cmul(g10, s0), cmul(g11, s1));
            }
            __syncthreads();
        }
        // fused controlled gates on ring (q -> q+1); 512 pairs each
        for (int q = 0; q < NQ; ++q) {
            const int t = (q + 1) % NQ;
            const float2 g00 = uc[c][q][0], g01 = uc[c][q][1];
            const float2 g10 = uc[c][q][2], g11 = uc[c][q][3];
            const int bc = NQ - 1 - q, bt = NQ - 1 - t;
            const int hi = bc > bt ? bc : bt, lo = bc > bt ? bt : bc;
            for (int p = tid; p < DIM / 4; p += 256) {
                int idx = ((p >> lo) << (lo + 1)) | (p & ((1 << lo) - 1));
                idx = ((idx >> hi) << (hi + 1)) | (idx & ((1 << hi) - 1));
                const int i0 = idx | (1 << bc);     // control bit set
                const int i1 = i0 | (1 << bt);
                const float2 s0 = st[i0], s1 = st[i1];
                st[i0] = cadd(cmul(g00, s0), cmul(g01, s1));
                st[i1] = cadd(cmul(g10, s0), cmul(g11, s1));
            }
            __syncthreads();
        }
    }

    // ---- expectation values: <X_q>, <Z_q> ----
    float xv[NQ], zv[NQ];
    #pragma unroll
    for (int q = 0; q < NQ; ++q) { xv[q] = 0.f; zv[q] = 0.f; }

    for (int i = tid; i < DIM; i += 256) {
        const float2 s = st[i];
        const float p2 = s.x * s.x + s.y * s.y;
        #pragma unroll
        for (int q = 0; q < NQ; ++q) {
            const int bp = NQ - 1 - q;
            if ((i >> bp) & 1) {
                zv[q] -= p2;
            } else {
                zv[q] += p2;
                const float2 t2 = st[i | (1 << bp)];
                xv[q] += 2.f * (s.x * t2.x + s.y * t2.y);  // 2*Re(conj(s)*t)
            }
        }
    }
    // wave32 reduction, then cross-wave via LDS float atomics (ds_add_f32)
    #pragma unroll
    for (int q = 0; q < NQ; ++q) {
        #pragma unroll
        for (int m = 16; m >= 1; m >>= 1) {
            xv[q] += __shfl_xor(xv[q], m, 32);
            zv[q] += __shfl_xor(zv[q], m, 32);
        }
    }
    if ((tid & 31) == 0) {
        #pragma unroll
        for (int q = 0; q < NQ; ++q) {
            atomicAdd(&acc[q], xv[q]);
            atomicAdd(&acc[NQ + q], zv[q]);
        }
    }
    __syncthreads();
    if (tid < NOUT) out[bb * NOUT + tid] = acc[tid];
}

// ============================================================
extern "C" void kernel_launch(void* const* d_in, const int* in_sizes, int n_in,
                              void* d_out, int out_size, void* d_ws, size_t ws_size,
                              hipStream_t stream) {
    (void)in_sizes; (void)n_in; (void)out_size;
    const float* x    = (const float*)d_in[0];  // [512][11]
    const float* W    = (const float*)d_in[1];  // [3][66][11]
    const float* bias = (const float*)d_in[2];  // [3][66]
    float* out        = (float*)d_out;          // [512][22]

    const size_t need = (size_t)BATCH * NANG * sizeof(float);
    const float* ang_pre = nullptr;
    if (d_ws && ws_size >= need) {
        float* ws = (float*)d_ws;
        dim3 grid(BATCH / 16, (NLIN + 15) / 16);   // 32 x 13 tiles
        angles_wmma_kernel<<<grid, 32, 0, stream>>>(x, W, bias, ws);
        ang_pre = ws;
    }
    qsim_kernel<<<BATCH, 256, 0, stream>>>(x, W, bias, ang_pre, out);
}